// GNHP_32796370273029
// MI455X (gfx1250) — compile-verified
//
#include <hip/hip_runtime.h>
#include <hip/hip_bf16.h>
#include <math.h>

// Problem constants (from reference)
#define BB    128          // batch
#define TP2   2050
#define TT    2049         // time steps
#define HH    64           // hidden
#define G7    448          // 7*H
#define KK    100          // num event types (output)
#define KE    103          // K+3 embedding rows
#define NROW  (BB * TT)    // 262272 == 16392 * 16 exactly

typedef float v2f __attribute__((ext_vector_type(2)));
typedef float v8f __attribute__((ext_vector_type(8)));
typedef int   v4i __attribute__((ext_vector_type(4)));

// ---------------------------------------------------------------------------
// WMMA f32 16x16x4 (fp32-exact; matches the fp32 reference)
// ---------------------------------------------------------------------------
__device__ __forceinline__ v8f wmma4(v2f a, v2f b, v8f c) {
    return __builtin_amdgcn_wmma_f32_16x16x4_f32(false, a, false, b, (short)0, c,
                                                 false, false);
}

// A fragment from a row-major LDS tile (row stride `stride`), cols k..k+3.
__device__ __forceinline__ v2f frag_a(const float* s, int stride, int k) {
    int lane = threadIdx.x & 31;
    int m  = lane & 15;
    int kk = k + ((lane >> 4) << 1);
    v2f a;
    a.x = s[m * stride + kk];
    a.y = s[m * stride + kk + 1];
    return a;
}

// B fragment from a row-major KxN source, rows k..k+3, cols n0..n0+15.
__device__ __forceinline__ v2f frag_b(const float* src, int ldn, int k, int n0) {
    int lane = threadIdx.x & 31;
    int n  = n0 + (lane & 15);
    int kk = k + ((lane >> 4) << 1);
    v2f b;
    b.x = src[(size_t)kk * ldn + n];
    b.y = src[(size_t)(kk + 1) * ldn + n];
    return b;
}

// Swizzled-layout B fragment: one aligned 8-byte LDS load per fragment.
// layout: s[((tile*16 + ks)*32 + lane)*2 + slot]
__device__ __forceinline__ v2f frag_b_sw(const float* s, int tile, int ks) {
    int lane = threadIdx.x & 31;
    return *(const v2f*)(s + (size_t)((tile * 16 + ks) * 32 + lane) * 2);
}

// ---------------------------------------------------------------------------
// Fast transcendentals (guarded gfx1250 trans ops)
// ---------------------------------------------------------------------------
__device__ __forceinline__ float rcp_(float x) {
#if __has_builtin(__builtin_amdgcn_rcpf)
    return __builtin_amdgcn_rcpf(x);
#else
    return 1.0f / x;
#endif
}
__device__ __forceinline__ float tanh_(float x) {
#if __has_builtin(__builtin_amdgcn_tanhf)
    return __builtin_amdgcn_tanhf(x);
#else
    float ax = fabsf(x);
    float e  = __expf(-2.0f * ax);
    float r  = (1.0f - e) * rcp_(1.0f + e);
    return copysignf(r, x);
#endif
}
__device__ __forceinline__ float sigmoid_(float x) {
    return rcp_(1.0f + __expf(-x));
}
__device__ __forceinline__ float softplus_(float x) {
    return (x > 20.0f) ? x : __logf(1.0f + __expf(x));
}

// ---------------------------------------------------------------------------
// Async global->LDS copy (16B per lane), with synchronous fallback.
// Builtin signature (from hipcc diagnostic): param 0 is `int __vector(4) *`.
// ---------------------------------------------------------------------------
__device__ __forceinline__ void async_cp16(const float* g, float* l) {
#if __has_builtin(__builtin_amdgcn_global_load_async_to_lds_b128)
    __builtin_amdgcn_global_load_async_to_lds_b128((v4i*)g, (v4i*)l, 0, 0);
#else
    *(float4*)l = *(const float4*)g;
#endif
}
__device__ __forceinline__ void wait_async0() {
#if __has_builtin(__builtin_amdgcn_s_wait_asynccnt)
    __builtin_amdgcn_s_wait_asynccnt(0);
#else
    asm volatile("s_wait_asynccnt 0x0" ::: "memory");
#endif
}

// ============================================================================
// Scan helpers
// ============================================================================
__device__ __forceinline__ v8f cinit(const float* sx, int tile) {
    int col = threadIdx.x & 15, half = (threadIdx.x >> 4) & 1;
    int n0 = tile * 16;
    v8f c;
    #pragma unroll
    for (int j = 0; j < 8; ++j) c[j] = sx[(j + half * 8) * G7 + n0 + col];
    return c;
}
__device__ __forceinline__ void gstore(float* sg, int tile, v8f c) {
    int col = threadIdx.x & 15, half = (threadIdx.x >> 4) & 1;
    int n0 = tile * 16;
    #pragma unroll
    for (int j = 0; j < 8; ++j) sg[(j + half * 8) * G7 + n0 + col] = c[j];
}

// ============================================================================
// Kernel (first in module so its disasm is visible): sequential CT-LSTM scan.
// 8 blocks x 16 batch rows, 256 threads. Swizzled Wh/h in LDS (one
// ds_load_b64 per fragment), k-outer A reuse, xproj gathers double-buffered
// via async global->LDS b128.
// ============================================================================
__global__ void __launch_bounds__(256)
scan_kernel(const int* __restrict__ event, const float* __restrict__ dtime,
            const float* __restrict__ Wh, const float* __restrict__ xproj,
            float* __restrict__ hs) {
    __shared__ __align__(16) float sWhS[28 * 16 * 64];   // 114688 B, B-frag swizzled
    __shared__ __align__(16) float sX[2][16 * G7];       // 2 x 28672 B gathered xproj
    __shared__ __align__(16) float sG[16 * G7];          // 28672 B
    __shared__ __align__(16) float sHS[16 * 64];         // A-frag swizzled h
    __shared__ __align__(16) float sC [16 * 64];
    __shared__ __align__(16) float sCb[16 * 64];
    __shared__ int   sEvPP[2][16];
    __shared__ float sDtPP[2][16];

    const int tid  = threadIdx.x;
    const int lane = tid & 31;
    const int wave = tid >> 5;
    const int b0   = blockIdx.x * 16;

    // ---- one-time staging: Wh into B-fragment swizzled layout ----
    for (int idx = tid; idx < 28 * 16 * 64; idx += 256) {
        int slot = idx & 1, ln = (idx >> 1) & 31, ks = (idx >> 6) & 15, tile = idx >> 10;
        int n = tile * 16 + (ln & 15);
        int k = ks * 4 + ((ln >> 4) << 1) + slot;
        sWhS[idx] = Wh[k * G7 + n];
    }
    for (int idx = tid; idx < 16 * 64; idx += 256) {
        sHS[idx] = 0.0f; sC[idx] = 0.0f; sCb[idx] = 0.0f;
    }
    if (tid < 16) {
        sEvPP[0][tid] = event[(size_t)(b0 + tid) * TP2 + 0];
        sEvPP[1][tid] = event[(size_t)(b0 + tid) * TP2 + 1];
        sDtPP[1][tid] = dtime[(size_t)(b0 + tid) * TP2 + 1];
    }
    __syncthreads();

    // ---- prologue: fill sX[0] with xproj rows for t = 0 ----
    {
        const int* ev0 = sEvPP[0];
        #pragma unroll
        for (int jj = 0; jj < 7; ++jj) {
            int cidx = wave * 224 + jj * 32 + lane;     // 1792 16B-chunks
            int m = cidx / 112, off = cidx - m * 112;
            async_cp16(xproj + (size_t)ev0[m] * G7 + off * 4,
                       &sX[0][m * G7 + off * 4]);
        }
    }
    wait_async0();
    __syncthreads();

    for (int t = 0; t < TT; ++t) {
        // ---- async prefetch of next step's xproj rows (overlaps WMMA) ----
        if (t + 1 < TT) {
            const int* evn = sEvPP[(t + 1) & 1];
            float* dst = sX[(t + 1) & 1];
            #pragma unroll
            for (int jj = 0; jj < 7; ++jj) {
                int cidx = wave * 224 + jj * 32 + lane;
                int m = cidx / 112, off = cidx - m * 112;
                async_cp16(xproj + (size_t)evn[m] * G7 + off * 4,
                           dst + m * G7 + off * 4);
            }
        }
        // ---- prefetch events/dt two steps ahead into the ping-pong slot ----
        if (tid < 16) {
            int ti = (t + 2 < TP2) ? (t + 2) : (TP2 - 1);
            sEvPP[t & 1][tid] = event[(size_t)(b0 + tid) * TP2 + ti];
            sDtPP[t & 1][tid] = dtime[(size_t)(b0 + tid) * TP2 + ti];
        }

        // ---- g = xproj[event_t] + h @ Wh : 28 tiles, k-outer, A reused ----
        const float* sx = sX[t & 1];
        if (wave < 4) {                       // wave-uniform: 4 tiles
            v8f c0 = cinit(sx, wave),      c1 = cinit(sx, wave + 8);
            v8f c2 = cinit(sx, wave + 16), c3 = cinit(sx, wave + 24);
            #pragma unroll
            for (int ks = 0; ks < 16; ++ks) {
                v2f a = *(const v2f*)(sHS + ks * 64 + lane * 2);
                c0 = wmma4(a, frag_b_sw(sWhS, wave,      ks), c0);
                c1 = wmma4(a, frag_b_sw(sWhS, wave + 8,  ks), c1);
                c2 = wmma4(a, frag_b_sw(sWhS, wave + 16, ks), c2);
                c3 = wmma4(a, frag_b_sw(sWhS, wave + 24, ks), c3);
            }
            gstore(sG, wave, c0);      gstore(sG, wave + 8,  c1);
            gstore(sG, wave + 16, c2); gstore(sG, wave + 24, c3);
        } else {                              // wave-uniform: 3 tiles
            v8f c0 = cinit(sx, wave), c1 = cinit(sx, wave + 8);
            v8f c2 = cinit(sx, wave + 16);
            #pragma unroll
            for (int ks = 0; ks < 16; ++ks) {
                v2f a = *(const v2f*)(sHS + ks * 64 + lane * 2);
                c0 = wmma4(a, frag_b_sw(sWhS, wave,      ks), c0);
                c1 = wmma4(a, frag_b_sw(sWhS, wave + 8,  ks), c1);
                c2 = wmma4(a, frag_b_sw(sWhS, wave + 16, ks), c2);
            }
            gstore(sG, wave, c0); gstore(sG, wave + 8, c1); gstore(sG, wave + 16, c2);
        }
        __syncthreads();

        // ---- gates: one (row, 4-col) group per thread, all b128 ----
        {
            const int r  = tid >> 4;
            const int j0 = (tid & 15) << 2;
            const float* gr = sG + r * G7;
            float4 xi  = *(const float4*)(gr + j0);
            float4 xf  = *(const float4*)(gr +  64 + j0);
            float4 xz  = *(const float4*)(gr + 128 + j0);
            float4 xo  = *(const float4*)(gr + 192 + j0);
            float4 xib = *(const float4*)(gr + 256 + j0);
            float4 xfb = *(const float4*)(gr + 320 + j0);
            float4 xd  = *(const float4*)(gr + 384 + j0);
            float4 cm  = *(const float4*)(sC  + r * 64 + j0);
            float4 cbp = *(const float4*)(sCb + r * 64 + j0);
            const float dtv = sDtPP[(t + 1) & 1][r];
            float4 cn, cbn, hn;
            #pragma unroll
            for (int q = 0; q < 4; ++q) {
                float i_  = sigmoid_((&xi.x)[q]);
                float f_  = sigmoid_((&xf.x)[q]);
                float z_  = tanh_   ((&xz.x)[q]);
                float o_  = sigmoid_((&xo.x)[q]);
                float ib_ = sigmoid_((&xib.x)[q]);
                float fb_ = sigmoid_((&xfb.x)[q]);
                float d_  = softplus_((&xd.x)[q]);
                float ci  = f_  * (&cm.x)[q]  + i_  * z_;
                float cbi = fb_ * (&cbp.x)[q] + ib_ * z_;
                float cnq = cbi + (ci - cbi) * __expf(-d_ * dtv);
                (&cn.x)[q]  = cnq;
                (&cbn.x)[q] = cbi;
                (&hn.x)[q]  = o_ * tanh_(cnq);
            }
            *(float4*)(sC  + r * 64 + j0) = cn;
            *(float4*)(sCb + r * 64 + j0) = cbn;
            // swizzled h for next step's A fragments (two b64 stores)
            v2f h01; h01.x = hn.x; h01.y = hn.y;
            v2f h23; h23.x = hn.z; h23.y = hn.w;
            *(v2f*)(sHS + (j0 >> 2) * 64 + r * 2)      = h01;
            *(v2f*)(sHS + (j0 >> 2) * 64 + 32 + r * 2) = h23;
            *(float4*)(hs + ((size_t)(b0 + r) * TT + t) * HH + j0) = hn;
        }
        wait_async0();
        __syncthreads();
    }
}

// ============================================================================
// Kernel: xproj[e,:] = in_emb[e,:] @ Wx + b   (103 x 448, L2-resident)
// ============================================================================
__global__ void __launch_bounds__(128)
xproj_kernel(const float* __restrict__ in_emb, const float* __restrict__ Wx,
             const float* __restrict__ bvec, float* __restrict__ xproj) {
    __shared__ __align__(16) float sE[16 * 65];
    const int tid = threadIdx.x;
    const int m0  = blockIdx.x * 16;

    for (int idx = tid; idx < 16 * 64; idx += 128) {
        int r = idx >> 6, k = idx & 63;
        int rg = m0 + r;
        sE[r * 65 + k] = (rg < KE) ? in_emb[rg * HH + k] : 0.0f;
    }
    __syncthreads();

    const int wave = tid >> 5;
    const int col  = tid & 15;
    const int half = (tid >> 4) & 1;
    for (int tile = wave; tile < 28; tile += 4) {
        const int n0 = tile * 16;
        v8f c = {};
        #pragma unroll
        for (int k = 0; k < HH; k += 4) {
            v2f a = frag_a(sE, 65, k);
            v2f b = frag_b(Wx, G7, k, n0);
            c = wmma4(a, b, c);
        }
        float bv = bvec[n0 + col];
        #pragma unroll
        for (int j = 0; j < 8; j++) {
            int row = m0 + j + half * 8;
            if (row < KE) xproj[(size_t)row * G7 + n0 + col] = c[j] + bv;
        }
    }
}

// ============================================================================
// Kernel: pre-swizzle out_emb^T into B-fragment layout (7 tiles x 16 ks)
// ============================================================================
__global__ void __launch_bounds__(256)
oeswz_kernel(const float* __restrict__ out_emb, float* __restrict__ oeS) {
    int idx = blockIdx.x * 256 + threadIdx.x;   // 7168 = 28 * 256
    int slot = idx & 1, ln = (idx >> 1) & 31, ks = (idx >> 6) & 15, tile = idx >> 10;
    int n = tile * 16 + (ln & 15);
    int k = ks * 4 + ((ln >> 4) << 1) + slot;
    oeS[idx] = (n < KK) ? out_emb[n * HH + k] : 0.0f;
}

// ============================================================================
// Kernel: out = softplus(hs @ out_emb^T). 16392 blocks x 128 threads,
// B operand pre-swizzled in global -> coalesced b128 staging.
// ============================================================================
__device__ __forceinline__ void store_out(float* out, int m0, int tile, v8f c) {
    int col = threadIdx.x & 15, half = (threadIdx.x >> 4) & 1;
    int n = tile * 16 + col;
    if (n < KK) {
        #pragma unroll
        for (int j = 0; j < 8; ++j)
            out[(size_t)(m0 + j + half * 8) * KK + n] = softplus_(c[j]);
    }
}

__global__ void __launch_bounds__(128)
logits_kernel(const float* __restrict__ hs, const float* __restrict__ oeS,
              float* __restrict__ out) {
    __shared__ __align__(16) float sHsS[16 * 64];       // A-frag swizzled
    __shared__ __align__(16) float sOeS[7 * 16 * 64];   // B-frag swizzled (28672 B)

    const int tid = threadIdx.x;
    const int lane = tid & 31;
    const int m0  = blockIdx.x * 16;

    for (int idx = tid; idx < 1024; idx += 128) {        // swizzle hs tile
        int slot = idx & 1, ln = (idx >> 1) & 31, ks = idx >> 6;
        int m = ln & 15, k = ks * 4 + ((ln >> 4) << 1) + slot;
        sHsS[idx] = hs[(size_t)(m0 + m) * HH + k];
    }
    for (int i = tid; i < 7168 / 4; i += 128)            // coalesced b128 copy
        ((float4*)sOeS)[i] = ((const float4*)oeS)[i];
    __syncthreads();

    const int wave = tid >> 5;
    if (wave < 3) {                                      // tiles {w, w+4}
        v8f c0 = {}, c1 = {};
        #pragma unroll
        for (int ks = 0; ks < 16; ++ks) {
            v2f a = *(const v2f*)(sHsS + ks * 64 + lane * 2);
            c0 = wmma4(a, frag_b_sw(sOeS, wave,     ks), c0);
            c1 = wmma4(a, frag_b_sw(sOeS, wave + 4, ks), c1);
        }
        store_out(out, m0, wave, c0);
        store_out(out, m0, wave + 4, c1);
    } else {                                             // tile {3}
        v8f c0 = {};
        #pragma unroll
        for (int ks = 0; ks < 16; ++ks) {
            v2f a = *(const v2f*)(sHsS + ks * 64 + lane * 2);
            c0 = wmma4(a, frag_b_sw(sOeS, 3, ks), c0);
        }
        store_out(out, m0, 3, c0);
    }
}

// ============================================================================
extern "C" void kernel_launch(void* const* d_in, const int* in_sizes, int n_in,
                              void* d_out, int out_size, void* d_ws, size_t ws_size,
                              hipStream_t stream) {
    const int*   event   = (const int*)  d_in[0];
    const float* dtime   = (const float*)d_in[1];
    const float* in_emb  = (const float*)d_in[2];
    const float* Wx      = (const float*)d_in[3];
    const float* Wh      = (const float*)d_in[4];
    const float* bvec    = (const float*)d_in[5];
    const float* out_emb = (const float*)d_in[6];
    float* out = (float*)d_out;

    // workspace: xproj (184576 B) | oeS (28672 B) | hs (NROW*64 f32)
    float* xproj = (float*)d_ws;
    float* oeS   = (float*)((char*)d_ws + 184576);
    float* hs    = (float*)((char*)d_ws + 184576 + 28672);

    xproj_kernel <<<(KE + 15) / 16, 128, 0, stream>>>(in_emb, Wx, bvec, xproj);
    oeswz_kernel <<<28,             256, 0, stream>>>(out_emb, oeS);
    scan_kernel  <<<BB / 16,        256, 0, stream>>>(event, dtime, Wh, xproj, hs);
    logits_kernel<<<NROW / 16,      128, 0, stream>>>(hs, oeS, out);
}